// MPSSSM_7713761264218
// MI455X (gfx1250) — compile-verified
//
#include <hip/hip_runtime.h>
#include <math.h>

typedef __attribute__((ext_vector_type(2))) float v2f;
typedef __attribute__((ext_vector_type(8))) float v8f;

#define B_SZ 32
#define SEQ 96
#define ENC_IN 21
#define D_MODEL 64
#define D_STATE 64
#define GATE_HID 128
#define GATE_OUT 8193
#define OUT_COLS 4032   // 2*PRED_LEN*ENC_IN
#define ROWS (B_SZ*SEQ) // 3072
#define GCOLS 8192

// padded row stride for 64x64 matrices in LDS: 68 words -> 16-lane A-frag
// reads map to 16 distinct banks (4*lane mod 64), rows stay 16B aligned.
#define SLD 68
#define SMAT (64*SLD)
#define IX(i,j) ((i)*SLD + (j))

__device__ __forceinline__ float gelu_f(float x) {
  return 0.5f * x * (1.0f + erff(x * 0.70710678118654752f));
}

// ---------------- CDNA5 async copy helpers --------------------------------
__device__ __forceinline__ void async_load_b128(unsigned lds_off,
                                                const float* gaddr) {
  asm volatile("global_load_async_to_lds_b128 %0, %1, off"
               :: "v"(lds_off), "v"(gaddr) : "memory");
}
__device__ __forceinline__ void wait_async0() {
  asm volatile("s_wait_asynccnt 0x0" ::: "memory");
}
__device__ __forceinline__ unsigned lds_off_of(const void* p) {
  return (unsigned)(unsigned long long)p; // flat->LDS uses addr[31:0]
}

// ---------------- WMMA f32 16x16x4 helpers -------------------------------
__device__ __forceinline__ v8f wmma4(v2f a, v2f b, v8f c) {
  return __builtin_amdgcn_wmma_f32_16x16x4_f32(false, a, false, b, (short)0, c,
                                               false, false);
}

// 64x64 matmul in LDS (row stride SLD), 8 waves, C = alpha*A@B
__device__ void mm64(const float* A, const float* B, float* C, float alpha) {
  __syncthreads();
  int wave = threadIdx.x >> 5;
  int lane = threadIdx.x & 31;
  int lrow = lane & 15;
  int khalf = (lane >> 4) << 1;
  int rbase_off = (lane >> 4) << 3;
#pragma unroll
  for (int i = 0; i < 2; ++i) {
    int tile = wave * 2 + i; // 16 tiles = 4x4
    int row0 = (tile >> 2) << 4;
    int col0 = (tile & 3) << 4;
    v8f acc = {0.f,0.f,0.f,0.f,0.f,0.f,0.f,0.f};
    for (int k0 = 0; k0 < 64; k0 += 4) {
      v2f a, b;
      int kk = k0 + khalf;
      a.x = A[(row0 + lrow) * SLD + kk];
      a.y = A[(row0 + lrow) * SLD + kk + 1];
      b.x = B[kk * SLD + col0 + lrow];
      b.y = B[(kk + 1) * SLD + col0 + lrow];
      acc = wmma4(a, b, acc);
    }
#pragma unroll
    for (int r = 0; r < 8; ++r)
      C[(row0 + rbase_off + r) * SLD + col0 + lrow] = alpha * acc[r];
  }
  __syncthreads();
}

// ---------------- Generic WMMA GEMM (small shapes) ------------------------
__global__ void __launch_bounds__(256)
gemm_wmma_kernel(const float* __restrict__ A, int lda,
                 const float* __restrict__ W, int ldw, int wcol,
                 const float* __restrict__ bias, int bcol,
                 float* __restrict__ C, int ldc,
                 int M, int N, int K, int act) {
  int wave = threadIdx.x >> 5;
  int gid = blockIdx.x * (blockDim.x >> 5) + wave;
  int tn = N >> 4;
  int tm = M >> 4;
  if (gid >= tm * tn) return; // wave-uniform exit
  int row0 = (gid / tn) << 4;
  int col0 = (gid % tn) << 4;
  int lane = threadIdx.x & 31;
  int lrow = lane & 15;
  int khalf = (lane >> 4) << 1;
  v8f acc = {0.f,0.f,0.f,0.f,0.f,0.f,0.f,0.f};
  for (int k0 = 0; k0 < K; k0 += 4) {
    int kk = k0 + khalf;
    v2f a, b;
    a.x = A[(size_t)(row0 + lrow) * lda + kk];
    a.y = A[(size_t)(row0 + lrow) * lda + kk + 1];
    b.x = W[(size_t)kk * ldw + wcol + col0 + lrow];
    b.y = W[(size_t)(kk + 1) * ldw + wcol + col0 + lrow];
    acc = wmma4(a, b, acc);
  }
  float bv = bias ? bias[bcol + col0 + lrow] : 0.0f;
  int rbase = row0 + ((lane >> 4) << 3);
#pragma unroll
  for (int r = 0; r < 8; ++r) {
    float v = acc[r] + bv;
    if (act == 1) v = gelu_f(v);
    else if (act == 2) v = 0.5f * tanhf(v);
    C[(size_t)(rbase + r) * ldc + col0 + lrow] = v;
  }
}

// ---------------- Big GEMM: LDS double-buffered + async loads -------------
#define KC 32
#define APAD 36
__global__ void __launch_bounds__(256)
gemm_big_kernel(const float* __restrict__ A, int lda,
                const float* __restrict__ W, int ldw,
                const float* __restrict__ bias,
                float* __restrict__ C, int ldc,
                int M, int N, int K) {
  __shared__ float sA[2][64 * APAD];
  __shared__ float sW[2][KC * 128];
  int nb = N >> 7;
  int row0 = (blockIdx.x / nb) << 6;
  int col0 = (blockIdx.x % nb) << 7;
  int tid = threadIdx.x;
  int wave = tid >> 5, lane = tid & 31;
  int lrow = lane & 15;
  int khalf = (lane >> 4) << 1;
  int rboff = (lane >> 4) << 3;
  int wm = (wave >> 2) << 5;
  int wn = (wave & 3) << 5;

  auto load_chunk = [&](int kc, int buf) {
#pragma unroll
    for (int v = 0; v < 2; ++v) { // A: 64 x 32
      int idx = tid * 2 + v;
      int ar = idx >> 3;
      int ac = (idx & 7) << 2;
      async_load_b128(lds_off_of(&sA[buf][ar * APAD + ac]),
                      A + (size_t)(row0 + ar) * lda + kc + ac);
    }
#pragma unroll
    for (int v = 0; v < 4; ++v) { // W: 32 x 128
      int idx = tid * 4 + v;
      int wr = idx >> 5;
      int wc = (idx & 31) << 2;
      async_load_b128(lds_off_of(&sW[buf][wr * 128 + wc]),
                      W + (size_t)(kc + wr) * ldw + col0 + wc);
    }
  };

  v8f zero = {0.f,0.f,0.f,0.f,0.f,0.f,0.f,0.f};
  v8f acc[2][2];
  acc[0][0] = zero; acc[0][1] = zero; acc[1][0] = zero; acc[1][1] = zero;

  load_chunk(0, 0);
  wait_async0();
  __syncthreads();
  int nchunks = K / KC;
  for (int c = 0; c < nchunks; ++c) {
    int buf = c & 1;
    if (c + 1 < nchunks) load_chunk((c + 1) * KC, buf ^ 1);
    const float* a_s = sA[buf];
    const float* w_s = sW[buf];
#pragma unroll
    for (int k0 = 0; k0 < KC; k0 += 4) {
      int kk = k0 + khalf;
      v2f afr[2], bfr[2];
#pragma unroll
      for (int i = 0; i < 2; ++i) {
        int r = wm + i * 16 + lrow;
        afr[i].x = a_s[r * APAD + kk];
        afr[i].y = a_s[r * APAD + kk + 1];
        int cc = wn + i * 16 + lrow;
        bfr[i].x = w_s[kk * 128 + cc];
        bfr[i].y = w_s[(kk + 1) * 128 + cc];
      }
#pragma unroll
      for (int i = 0; i < 2; ++i)
#pragma unroll
        for (int j = 0; j < 2; ++j)
          acc[i][j] = wmma4(afr[i], bfr[j], acc[i][j]);
    }
    __syncthreads();
    if (c + 1 < nchunks) {
      wait_async0();
      __syncthreads();
    }
  }
#pragma unroll
  for (int i = 0; i < 2; ++i) {
    int rbase = row0 + wm + i * 16 + rboff;
#pragma unroll
    for (int j = 0; j < 2; ++j) {
      int cc = col0 + wn + j * 16 + lrow;
      float bv = bias[cc];
#pragma unroll
      for (int r = 0; r < 8; ++r)
        C[(size_t)(rbase + r) * ldc + cc] = 0.5f * tanhf(acc[i][j][r] + bv);
    }
  }
}

// ---------------- repack gW3 cols 1..8192 into aligned buffer -------------
__global__ void repack_gw3_kernel(const float* __restrict__ gW3,
                                  float* __restrict__ Wp) {
  int i = blockIdx.x * blockDim.x + threadIdx.x;
  if (i >= GATE_HID * GCOLS) return;
  int k = i >> 13, j = i & (GCOLS - 1);
  Wp[i] = gW3[(size_t)k * GATE_OUT + 1 + j];
}

// ---------------- embed + LayerNorm --------------------------------------
__global__ void __launch_bounds__(64)
embed_ln_kernel(const float* __restrict__ x, const float* __restrict__ W_emb,
                const float* __restrict__ b_emb, const float* __restrict__ ln_g,
                const float* __restrict__ ln_b, float* __restrict__ H) {
  int row = blockIdx.x;
  int j = threadIdx.x;
  __shared__ float se[64];
  __shared__ float sred[2];
  float acc = b_emb[j];
  const float* xr = x + row * ENC_IN;
  for (int c = 0; c < ENC_IN; ++c) acc += xr[c] * W_emb[c * D_MODEL + j];
  se[j] = acc;
  __syncthreads();
  if (j == 0) {
    float mu = 0.f;
    for (int k = 0; k < 64; ++k) mu += se[k];
    mu *= (1.0f / 64.0f);
    float va = 0.f;
    for (int k = 0; k < 64; ++k) { float d = se[k] - mu; va += d * d; }
    va *= (1.0f / 64.0f);
    sred[0] = mu;
    sred[1] = rsqrtf(va + 1e-5f);
  }
  __syncthreads();
  H[row * D_MODEL + j] = (acc - sred[0]) * sred[1] * ln_g[j] + ln_b[j];
}

// ---------------- delta column (gate output col 0) ------------------------
__global__ void delta_kernel(const float* __restrict__ T2,
                             const float* __restrict__ gW3,
                             const float* __restrict__ gb3,
                             float* __restrict__ delta_raw) {
  int row = blockIdx.x * blockDim.x + threadIdx.x;
  if (row >= ROWS) return;
  float a = gb3[0];
  const float* tr = T2 + (size_t)row * GATE_HID;
  for (int k = 0; k < GATE_HID; ++k) a += tr[k] * gW3[(size_t)k * GATE_OUT];
  float sp = (a > 20.0f) ? a : log1pf(expf(a));
  float d = sp + 0.01f;
  delta_raw[row] = fminf(fmaxf(d, 0.01f), 1.0f);
}

// ---------------- sequential scan: one workgroup per batch element --------
__global__ void __launch_bounds__(256)
scan_kernel(const float* __restrict__ H, const float* __restrict__ G,
            const float* __restrict__ delta_raw, const float* __restrict__ eps,
            const float* __restrict__ A_base,
            const float* __restrict__ emW1, const float* __restrict__ emb1,
            const float* __restrict__ emW2, const float* __restrict__ emb2,
            const float* __restrict__ evW1, const float* __restrict__ evb1,
            const float* __restrict__ evW2, const float* __restrict__ evb2,
            const float* __restrict__ prW, const float* __restrict__ prb,
            const float* __restrict__ pr_logvar, float* __restrict__ finalLat) {
  extern __shared__ float smem[];
  float* sX = smem;               // 64xSLD
  float* sT = smem + SMAT;        // 64xSLD
  float* sE = smem + 2 * SMAT;    // 64xSLD (A_bar)
  float* sPhi = smem + 3 * SMAT;  // 64xSLD (phi1)
  float* sS1 = smem + 4 * SMAT;   // 64xSLD scratch
  float* sG = smem + 5 * SMAT;    // 4096, async-prefetched G row (Bm half)
  __shared__ float sstate[64], sh[64], scat[128], shid[64];
  __shared__ float smq[64], slq[64], smp[64], sss[64];
  __shared__ float sred[64], svv[64], sww[64];
  __shared__ float spart[256];
  __shared__ float sscal[4];
  __shared__ int s_s;

  int b = blockIdx.x;
  int tid = threadIdx.x;
  int out = tid & 63, part = tid >> 6; // 4-way K-split for matvecs
  if (tid < 64) sstate[tid] = 0.0f;
  __syncthreads();

  for (int t = 0; t < SEQ; ++t) {
    int row = b * SEQ + t;
    const float* Grow = G + (size_t)row * GCOLS;

    // async prefetch of Bm half of the G row: consumed after expm
    {
      const float* gsrc = Grow + tid * 16;
      unsigned lo = lds_off_of(&sG[tid * 16]);
#pragma unroll
      for (int v = 0; v < 4; ++v)
        async_load_b128(lo + 16u * v, gsrc + 4 * v);
    }

    if (tid < 64) {
      float hv = H[row * 64 + tid];
      sh[tid] = hv;
      scat[tid] = sstate[tid];
      scat[64 + tid] = hv;
    }
    __syncthreads();

    // --- encoder MLPs: 256-thread matvecs (4 K-partials per output) ---
    { // h1 = gelu(cat @ emW1 + emb1), K=128
      float a = 0.f;
      const float* wp = emW1 + out;
      for (int k = part * 32; k < part * 32 + 32; ++k) a += scat[k] * wp[k * 64];
      spart[tid] = a;
    }
    __syncthreads();
    if (tid < 64)
      shid[tid] = gelu_f(emb1[tid] + spart[tid] + spart[64 + tid] +
                         spart[128 + tid] + spart[192 + tid]);
    __syncthreads();
    { // mq = h1 @ emW2 + emb2, K=64
      float a = 0.f;
      const float* wp = emW2 + out;
      for (int k = part * 16; k < part * 16 + 16; ++k) a += shid[k] * wp[k * 64];
      spart[tid] = a;
    }
    __syncthreads();
    if (tid < 64)
      smq[tid] = emb2[tid] + spart[tid] + spart[64 + tid] + spart[128 + tid] +
                 spart[192 + tid];
    __syncthreads();
    { // h2 = gelu(cat @ evW1 + evb1), K=128
      float a = 0.f;
      const float* wp = evW1 + out;
      for (int k = part * 32; k < part * 32 + 32; ++k) a += scat[k] * wp[k * 64];
      spart[tid] = a;
    }
    __syncthreads();
    if (tid < 64)
      shid[tid] = gelu_f(evb1[tid] + spart[tid] + spart[64 + tid] +
                         spart[128 + tid] + spart[192 + tid]);
    __syncthreads();
    { // lq = clamp(h2 @ evW2 + evb2), K=64
      float a = 0.f;
      const float* wp = evW2 + out;
      for (int k = part * 16; k < part * 16 + 16; ++k) a += shid[k] * wp[k * 64];
      spart[tid] = a;
    }
    __syncthreads();
    if (tid < 64)
      slq[tid] = fminf(fmaxf(evb2[tid] + spart[tid] + spart[64 + tid] +
                                 spart[128 + tid] + spart[192 + tid],
                             -8.0f), 8.0f);
    __syncthreads();
    { // mp = state @ prW + prb, K=64
      float a = 0.f;
      const float* wp = prW + out;
      for (int k = part * 16; k < part * 16 + 16; ++k) a += sstate[k] * wp[k * 64];
      spart[tid] = a;
    }
    __syncthreads();
    if (tid < 64) {
      smp[tid] = prb[tid] + spart[tid] + spart[64 + tid] + spart[128 + tid] +
                 spart[192 + tid];
      float ss = smq[tid] + expf(0.5f * slq[tid]) * eps[((size_t)t * B_SZ + b) * 64 + tid];
      sss[tid] = ss;
      float lp = pr_logvar[tid];
      float dd = smq[tid] - smp[tid];
      sred[tid] = 0.5f * (lp - slq[tid] + (expf(slq[tid]) + dd * dd) / expf(lp) - 1.0f);
    }
    __syncthreads();
    if (tid == 0) {
      float rate = 0.f;
      for (int k = 0; k < 64; ++k) rate += sred[k];
      float fb = rate * (1.0f / 64.0f);
      float inv = 1.0f / (1.0f + 0.2f * fb);
      sscal[0] = fb;
      sscal[1] = delta_raw[row] * inv;
      sscal[2] = inv;
    }
    __syncthreads();
    float delta = sscal[1], mscale = sscal[2];

    // --- M = A_base * delta (padded layout), inf-norm, scaling exponent ---
    for (int e = tid; e < 4096; e += 256) {
      int i = e >> 6, j = e & 63;
      sT[IX(i, j)] = A_base[e] * delta;
    }
    __syncthreads();
    { // row inf-norm partials
      float a = 0.f;
      for (int j = part * 16; j < part * 16 + 16; ++j) a += fabsf(sT[IX(out, j)]);
      spart[tid] = a;
    }
    __syncthreads();
    if (tid < 64)
      sred[tid] = spart[tid] + spart[64 + tid] + spart[128 + tid] + spart[192 + tid];
    __syncthreads();
    if (tid == 0) {
      float nrm = 0.f;
      for (int k = 0; k < 64; ++k) nrm = fmaxf(nrm, sred[k]);
      int s = 0;
      if (nrm > 0.5f) s = (int)ceilf(log2f(nrm * 2.0f));
      if (s < 0) s = 0;
      if (s > 24) s = 24;
      s_s = s;
      sscal[3] = ldexpf(1.0f, -s);
    }
    __syncthreads();
    int s = s_s;
    float inv2s = sscal[3];

    // --- Taylor: E = e^X, Phi = phi1(X), X = M/2^s, order 8 ---
    for (int e = tid; e < 4096; e += 256) {
      int i = e >> 6, j = e & 63;
      float xv = sT[IX(i, j)] * inv2s;
      float dg = (i == j) ? 1.0f : 0.0f;
      sX[IX(i, j)] = xv;
      sE[IX(i, j)] = dg + xv;
      sPhi[IX(i, j)] = dg + 0.5f * xv;
      sT[IX(i, j)] = xv;
    }
    for (int k = 2; k <= 8; ++k) {
      mm64(sT, sX, sS1, 1.0f / (float)k);
      float invk1 = 1.0f / (float)(k + 1);
      for (int e = tid; e < 4096; e += 256) {
        int i = e >> 6, j = e & 63;
        float tv = sS1[IX(i, j)];
        sT[IX(i, j)] = tv;
        sE[IX(i, j)] += tv;
        sPhi[IX(i, j)] += tv * invk1;
      }
    }
    __syncthreads();
    // --- squaring: E <- E^2, Phi <- Phi*(E+I)/2 ---
    for (int it = 0; it < s; ++it) {
      for (int e = tid; e < 4096; e += 256) {
        int i = e >> 6, j = e & 63;
        float dg = (i == j) ? 1.0f : 0.0f;
        sS1[IX(i, j)] = 0.5f * (sE[IX(i, j)] + dg);
      }
      mm64(sPhi, sS1, sT, 1.0f);
      for (int e = tid; e < 4096; e += 256) {
        int i = e >> 6, j = e & 63;
        sPhi[IX(i, j)] = sT[IX(i, j)];
      }
      mm64(sE, sE, sS1, 1.0f);
      for (int e = tid; e < 4096; e += 256) {
        int i = e >> 6, j = e & 63;
        sE[IX(i, j)] = sS1[IX(i, j)];
      }
      __syncthreads();
    }
    // --- B_bar = Phi @ (Bm * d): consume async-prefetched G row -----------
    wait_async0();
    __syncthreads();
    float nd = mscale * delta;
    for (int e = tid; e < 4096; e += 256) {
      int i = e >> 6, j = e & 63;
      sX[IX(i, j)] = sG[e] * nd;
    }
    mm64(sPhi, sX, sT, 1.0f); // B_bar in sT

    // --- spectral norm of A_bar (sE) via power iteration (256-thread) ---
    if (tid < 64) svv[tid] = 0.125f;
    __syncthreads();
    for (int it = 0; it < 16; ++it) {
      { // w = E v
        float a = 0.f;
        for (int j = part * 16; j < part * 16 + 16; ++j) a += sE[IX(out, j)] * svv[j];
        spart[tid] = a;
      }
      __syncthreads();
      if (tid < 64)
        sww[tid] = spart[tid] + spart[64 + tid] + spart[128 + tid] + spart[192 + tid];
      __syncthreads();
      { // u = E^T w
        float a = 0.f;
        for (int j = part * 16; j < part * 16 + 16; ++j) a += sE[IX(j, out)] * sww[j];
        spart[tid] = a;
      }
      __syncthreads();
      if (tid < 64)
        sred[tid] = spart[tid] + spart[64 + tid] + spart[128 + tid] + spart[192 + tid];
      __syncthreads();
      if (tid == 0) {
        float n2 = 0.f;
        for (int k = 0; k < 64; ++k) n2 += sred[k] * sred[k];
        sscal[3] = rsqrtf(n2 + 1e-30f);
      }
      __syncthreads();
      if (tid < 64) svv[tid] = sred[tid] * sscal[3];
      __syncthreads();
    }
    { // w = E v, sn = ||w||
      float a = 0.f;
      for (int j = part * 16; j < part * 16 + 16; ++j) a += sE[IX(out, j)] * svv[j];
      spart[tid] = a;
    }
    __syncthreads();
    if (tid < 64)
      sww[tid] = spart[tid] + spart[64 + tid] + spart[128 + tid] + spart[192 + tid];
    __syncthreads();
    if (tid == 0) {
      float n2 = 0.f;
      for (int k = 0; k < 64; ++k) n2 += sww[k] * sww[k];
      float sn = sqrtf(n2);
      sscal[3] = fminf(0.999f / (sn + 1e-12f), 1.0f);
    }
    __syncthreads();
    float ascale = sscal[3];

    // --- state update: new = ascale*A_bar@ss + B_bar@h (256-thread) ---
    {
      float a = 0.f, bb = 0.f;
      for (int j = part * 16; j < part * 16 + 16; ++j) {
        a += sE[IX(out, j)] * sss[j];
        bb += sT[IX(out, j)] * sh[j];
      }
      spart[tid] = ascale * a + bb;
    }
    __syncthreads();
    if (tid < 64)
      svv[tid] = spart[tid] + spart[64 + tid] + spart[128 + tid] + spart[192 + tid];
    __syncthreads();
    if (tid < 64) sstate[tid] = svv[tid];
    if (t == SEQ - 1) {
      float a = 0.f;
      const float* Crow = Grow + 4096 + out * 64;
      for (int j = part * 16; j < part * 16 + 16; ++j) a += Crow[j] * sss[j];
      spart[tid] = a;
      __syncthreads();
      if (tid < 64)
        finalLat[b * 64 + tid] = (spart[tid] + spart[64 + tid] +
                                  spart[128 + tid] + spart[192 + tid]) * mscale;
    }
    __syncthreads();
  }
}

// ---------------- head hidden: gelu(final @ hW1 + hb1) --------------------
__global__ void head_hidden_kernel(const float* __restrict__ finalLat,
                                   const float* __restrict__ hW1,
                                   const float* __restrict__ hb1,
                                   float* __restrict__ hidden) {
  int i = blockIdx.x * blockDim.x + threadIdx.x;
  if (i >= B_SZ * D_MODEL) return;
  int bb = i >> 6, j = i & 63;
  float a = hb1[j];
  const float* fr = finalLat + bb * 64;
  for (int k = 0; k < 64; ++k) a += fr[k] * hW1[k * 64 + j];
  hidden[i] = gelu_f(a);
}

extern "C" void kernel_launch(void* const* d_in, const int* in_sizes, int n_in,
                              void* d_out, int out_size, void* d_ws,
                              size_t ws_size, hipStream_t stream) {
  (void)in_sizes; (void)n_in; (void)out_size; (void)ws_size;
  const float* x = (const float*)d_in[0];
  const float* eps = (const float*)d_in[1];
  const float* A_base = (const float*)d_in[2];
  const float* W_emb = (const float*)d_in[3];
  const float* b_emb = (const float*)d_in[4];
  const float* ln_g = (const float*)d_in[5];
  const float* ln_b = (const float*)d_in[6];
  const float* gW1 = (const float*)d_in[7];
  const float* gb1 = (const float*)d_in[8];
  const float* gW2 = (const float*)d_in[9];
  const float* gb2 = (const float*)d_in[10];
  const float* gW3 = (const float*)d_in[11];
  const float* gb3 = (const float*)d_in[12];
  const float* emW1 = (const float*)d_in[13];
  const float* emb1 = (const float*)d_in[14];
  const float* emW2 = (const float*)d_in[15];
  const float* emb2 = (const float*)d_in[16];
  const float* evW1 = (const float*)d_in[17];
  const float* evb1 = (const float*)d_in[18];
  const float* evW2 = (const float*)d_in[19];
  const float* evb2 = (const float*)d_in[20];
  const float* prW = (const float*)d_in[21];
  const float* prb = (const float*)d_in[22];
  const float* pr_logvar = (const float*)d_in[23];
  const float* hW1 = (const float*)d_in[24];
  const float* hb1 = (const float*)d_in[25];
  const float* hW2 = (const float*)d_in[26];
  const float* hb2 = (const float*)d_in[27];

  float* ws = (float*)d_ws;
  float* H = ws;                        // 3072*64
  float* T1 = H + ROWS * 64;            // 3072*128
  float* T2 = T1 + ROWS * 128;          // 3072*128
  float* deltaR = T2 + ROWS * 128;      // 3072
  float* finalL = deltaR + ROWS;        // 2048
  float* hidden = finalL + B_SZ * 64;   // 2048
  float* Wp = hidden + B_SZ * 64;       // 128*8192
  float* G = Wp + GATE_HID * GCOLS;     // 3072*8192 (~96 MB)

  embed_ln_kernel<<<ROWS, 64, 0, stream>>>(x, W_emb, b_emb, ln_g, ln_b, H);

  auto launch_gemm = [&](const float* A, int lda, const float* W, int ldw,
                         int wcol, const float* bias, int bcol, float* C,
                         int ldc, int M, int N, int K, int act) {
    int tiles = (M / 16) * (N / 16);
    int blocks = (tiles + 7) / 8;
    gemm_wmma_kernel<<<blocks, 256, 0, stream>>>(A, lda, W, ldw, wcol, bias,
                                                 bcol, C, ldc, M, N, K, act);
  };

  launch_gemm(H, 64, gW1, GATE_HID, 0, gb1, 0, T1, GATE_HID, ROWS, GATE_HID, 64, 1);
  launch_gemm(T1, GATE_HID, gW2, GATE_HID, 0, gb2, 0, T2, GATE_HID, ROWS, GATE_HID, GATE_HID, 1);

  repack_gw3_kernel<<<(GATE_HID * GCOLS + 255) / 256, 256, 0, stream>>>(gW3, Wp);
  {
    int blocks = (ROWS / 64) * (GCOLS / 128); // 3072
    gemm_big_kernel<<<blocks, 256, 0, stream>>>(T2, GATE_HID, Wp, GCOLS,
                                                gb3 + 1, G, GCOLS,
                                                ROWS, GCOLS, GATE_HID);
  }
  delta_kernel<<<(ROWS + 255) / 256, 256, 0, stream>>>(T2, gW3, gb3, deltaR);

  scan_kernel<<<B_SZ, 256, (5 * SMAT + 4096) * sizeof(float), stream>>>(
      H, G, deltaR, eps, A_base, emW1, emb1, emW2, emb2, evW1, evb1, evW2,
      evb2, prW, prb, pr_logvar, finalL);

  head_hidden_kernel<<<(B_SZ * 64 + 255) / 256, 256, 0, stream>>>(finalL, hW1, hb1, hidden);
  launch_gemm(hidden, 64, hW2, OUT_COLS, 0, hb2, 0, (float*)d_out, OUT_COLS,
              B_SZ, OUT_COLS, 64, 0);
}